// AlphaFold3Loss_49701361550140
// MI455X (gfx1250) — compile-verified
//
#include <hip/hip_runtime.h>
#include <math.h>

#define NT 768
#define NA 3072
#define NBINS 64
#define SLOT 68   // 64 logits + 4 floats pad: 272B = 17*16B, kills LDS bank conflicts

typedef __attribute__((ext_vector_type(2))) float v2f;
typedef __attribute__((ext_vector_type(8))) float v8f;

__device__ __forceinline__ float sigm(float x){ return 1.0f/(1.0f+expf(-x)); }

// ws layout (floats):
// 0: distogram masked error sum
// 1: lddt numerator   2: lddt denominator
// 3: Sw   4-6: Sum(w*gt)   7-9: Sum(w*pred)   10-18: Sum(w*pred_i*gt_j)
// 19-27: R   28-30: mu(gt)   31-33: mu(pred)
// 34: mse acc   35: mask sum

// ------------------------------------------------------------------
// Distogram loss: one block per 16x16 (i,j) tile.
// Wave 0 computes the pairwise-distance tile via V_WMMA_F32_16X16X4_F32.
// Each thread stages its pair's 64 logits into LDS with
// GLOBAL_LOAD_ASYNC_TO_LDS_B128 (ASYNCcnt), then does the log-softmax
// out of LDS (two passes + bin gather, all LDS-resident).
// ------------------------------------------------------------------
__global__ __launch_bounds__(256)
void k_disto(const float* __restrict__ logits, const float* __restrict__ pos,
             const float* __restrict__ tm, float* __restrict__ ws)
{
  __shared__ float sLog[256*SLOT];          // ~69.6 KB staged logits tile
  __shared__ float sPi[16][4], sPj[16][4];
  __shared__ float sNi[16], sNj[16];
  __shared__ float sD2[256];
  __shared__ float red[256];
  const int t  = threadIdx.x;
  const int i0 = blockIdx.x*16, j0 = blockIdx.y*16;

  // ---- kick off async staging of this block's 64KB logits tile ----
  {
    const int il = t>>4, jl = t&15;
    const float* lg = logits + ((size_t)(i0+il)*NT + (j0+jl))*NBINS;
    __builtin_prefetch(lg, 0, 1);
    unsigned lds_base = (unsigned)(uintptr_t)(void*)&sLog[t*SLOT];
    unsigned long long gbase = (unsigned long long)(uintptr_t)lg;
    // async load semantics add INST_OFFSET to both LDS and global addresses,
    // so fixed bases + offset:k*16 walk both sides together.
    #pragma unroll
    for (int k=0;k<16;k++){
      asm volatile("global_load_async_to_lds_b128 %0, %1, off offset:%2"
                   :: "v"(lds_base), "v"(gbase), "i"(k*16) : "memory");
    }
  }

  if (t < 16){
    const float* p = pos + ((size_t)(i0+t)*4 + 1)*3;   // CA_POS = 1
    float x=p[0], y=p[1], z=p[2];
    sPi[t][0]=x; sPi[t][1]=y; sPi[t][2]=z; sPi[t][3]=0.f;
    sNi[t]=x*x+y*y+z*z;
  } else if (t < 32){
    int r = t-16;
    const float* p = pos + ((size_t)(j0+r)*4 + 1)*3;
    float x=p[0], y=p[1], z=p[2];
    sPj[r][0]=x; sPj[r][1]=y; sPj[r][2]=z; sPj[r][3]=0.f;
    sNj[r]=x*x+y*y+z*z;
  }
  __syncthreads();

  if (t < 32){
    const int m = t & 15; const bool hi = t >= 16;
    // A 16x4: lanes 0-15 K={0,1}, lanes 16-31 K={2,3}; B 4x16 mirrored.
    v2f a, b;
    a.x = hi ? sPi[m][2] : sPi[m][0];
    a.y = hi ? sPi[m][3] : sPi[m][1];
    b.x = hi ? sPj[m][2] : sPj[m][0];
    b.y = hi ? sPj[m][3] : sPj[m][1];
    v8f c = {};
    c = __builtin_amdgcn_wmma_f32_16x16x4_f32(false, a, false, b, (short)0, c, false, false);
    #pragma unroll
    for (int v=0; v<8; ++v){
      int il = v + (hi ? 8 : 0);                       // C/D layout: M = v + 8*(lane>=16)
      sD2[il*16 + m] = sNi[il] + sNj[m] - 2.0f*c[v];
    }
  }
  __syncthreads();

  const int il = t>>4, jl = t&15;
  const int i = i0+il, j = j0+jl;
  float d2 = sD2[t];
  float s  = sqrtf(fmaxf(d2, 0.f));
  // boundaries_k = (32k/62)^2, k=0..62 ; bin = #{k : d2 > b_k} = clamp(ceil(s*62/32),0,63)
  float tb = ceilf(s * (62.0f/32.0f));
  int bin = (tb >= 63.f) ? 63 : (int)tb;
  if (bin < 0) bin = 0;

  // wait only for our own async staging (each thread reads its own slot)
  asm volatile("s_wait_asynccnt 0x0" ::: "memory");
  const float* my = &sLog[t*SLOT];

  float mx = -3.4e38f;
  #pragma unroll
  for (int k=0;k<NBINS;k+=4){
    float4 vv = *(const float4*)(my + k);
    mx = fmaxf(mx, fmaxf(fmaxf(vv.x, vv.y), fmaxf(vv.z, vv.w)));
  }
  float se = 0.f;
  #pragma unroll
  for (int k=0;k<NBINS;k+=4){
    float4 vv = *(const float4*)(my + k);
    se += expf(vv.x-mx)+expf(vv.y-mx)+expf(vv.z-mx)+expf(vv.w-mx);
  }
  float err = -(my[bin] - mx - logf(se));
  err *= tm[i]*tm[j];

  red[t] = err; __syncthreads();
  for (int st=128; st>0; st>>=1){ if (t<st) red[t]+=red[t+st]; __syncthreads(); }
  if (t==0) atomicAdd(&ws[0], red[0]);
}

// ------------------------------------------------------------------
// Smooth LDDT: one block per 16x16 atom-pair tile; two WMMAs per tile
// (pred-Gram and gt-Gram), then per-pair sigmoids.
// ------------------------------------------------------------------
__global__ __launch_bounds__(256)
void k_lddt(const float* __restrict__ pred, const float* __restrict__ gt,
            const float* __restrict__ ae, float* __restrict__ ws)
{
  __shared__ float sPi[16][4], sPj[16][4], sGi[16][4], sGj[16][4];
  __shared__ float nPi[16], nPj[16], nGi[16], nGj[16];
  __shared__ float sDP[256], sDG[256];
  __shared__ float red[256];
  const int t  = threadIdx.x;
  const int i0 = blockIdx.x*16, j0 = blockIdx.y*16;

  if (t < 64){
    int which = t>>4, r = t&15;
    const float* src = (which < 2) ? pred : gt;
    int row = (which & 1) ? (j0+r) : (i0+r);
    const float* p = src + (size_t)row*3;
    float x=p[0], y=p[1], z=p[2];
    float (*dst)[4] = (which==0)? sPi : (which==1)? sPj : (which==2)? sGi : sGj;
    float* nd       = (which==0)? nPi : (which==1)? nPj : (which==2)? nGi : nGj;
    dst[r][0]=x; dst[r][1]=y; dst[r][2]=z; dst[r][3]=0.f;
    nd[r]=x*x+y*y+z*z;
  }
  __syncthreads();

  if (t < 32){
    const int m = t & 15; const bool hi = t >= 16;
    v2f ap,bp,ag,bg;
    ap.x = hi? sPi[m][2]:sPi[m][0]; ap.y = hi? sPi[m][3]:sPi[m][1];
    bp.x = hi? sPj[m][2]:sPj[m][0]; bp.y = hi? sPj[m][3]:sPj[m][1];
    ag.x = hi? sGi[m][2]:sGi[m][0]; ag.y = hi? sGi[m][3]:sGi[m][1];
    bg.x = hi? sGj[m][2]:sGj[m][0]; bg.y = hi? sGj[m][3]:sGj[m][1];
    v8f cz = {};
    v8f cp = __builtin_amdgcn_wmma_f32_16x16x4_f32(false, ap, false, bp, (short)0, cz, false, false);
    v8f cg = __builtin_amdgcn_wmma_f32_16x16x4_f32(false, ag, false, bg, (short)0, cz, false, false);
    #pragma unroll
    for (int v=0; v<8; ++v){
      int il = v + (hi?8:0);
      sDP[il*16+m] = cp[v];
      sDG[il*16+m] = cg[v];
    }
  }
  __syncthreads();

  const int il=t>>4, jl=t&15;
  const int i=i0+il, j=j0+jl;
  float dp = sqrtf(fmaxf(nPi[il]+nPj[jl]-2.f*sDP[t], 0.f) + 1e-6f);
  float dg = sqrtf(fmaxf(nGi[il]+nGj[jl]-2.f*sDG[t], 0.f) + 1e-6f);
  float delta = fabsf(dg-dp);
  float el = 0.25f*(sigm(0.5f-delta)+sigm(1.f-delta)+sigm(2.f-delta)+sigm(4.f-delta));
  float c  = (dg < 15.f ? 1.f : 0.f) * ae[i]*ae[j] * ((i==j)?0.f:1.f);

  red[t] = el*c; __syncthreads();
  for (int st=128; st>0; st>>=1){ if (t<st) red[t]+=red[t+st]; __syncthreads(); }
  if (t==0) atomicAdd(&ws[1], red[0]);
  __syncthreads();
  red[t] = c; __syncthreads();
  for (int st=128; st>0; st>>=1){ if (t<st) red[t]+=red[t+st]; __syncthreads(); }
  if (t==0) atomicAdd(&ws[2], red[0]);
}

// ------------------------------------------------------------------
// Rigid-align statistics: Sw, Sum(w*gt), Sum(w*pred), Sum(w*pred_i*gt_j)
// ------------------------------------------------------------------
__global__ __launch_bounds__(256)
void k_astats(const float* __restrict__ pred, const float* __restrict__ gt,
              const float* __restrict__ ae, float* __restrict__ ws)
{
  __shared__ float red[256];
  const int t = threadIdx.x;
  const int a = blockIdx.x*256 + t;
  float m = ae[a];
  float w = m*m;                 // weights * mask, both = atom_exists
  float p0=pred[a*3+0], p1=pred[a*3+1], p2=pred[a*3+2];
  float g0=gt[a*3+0],   g1=gt[a*3+1],   g2=gt[a*3+2];
  float vals[16];
  vals[0]=w;
  vals[1]=w*g0; vals[2]=w*g1; vals[3]=w*g2;
  vals[4]=w*p0; vals[5]=w*p1; vals[6]=w*p2;
  vals[7] =w*p0*g0; vals[8] =w*p0*g1; vals[9] =w*p0*g2;
  vals[10]=w*p1*g0; vals[11]=w*p1*g1; vals[12]=w*p1*g2;
  vals[13]=w*p2*g0; vals[14]=w*p2*g1; vals[15]=w*p2*g2;
  for (int k=0;k<16;k++){
    red[t]=vals[k]; __syncthreads();
    for (int st=128; st>0; st>>=1){ if (t<st) red[t]+=red[t+st]; __syncthreads(); }
    if (t==0) atomicAdd(&ws[3+k], red[0]);
    __syncthreads();
  }
}

// ------------------------------------------------------------------
// 3x3 Kabsch via Jacobi eigen-decomposition of H^T H (single thread)
// ------------------------------------------------------------------
__global__ void k_svd(float* __restrict__ ws)
{
  if (threadIdx.x != 0 || blockIdx.x != 0) return;
  float Sw = ws[3];
  float Sg[3] = {ws[4],ws[5],ws[6]};
  float Sp[3] = {ws[7],ws[8],ws[9]};
  float Spg[9]; for (int k=0;k<9;k++) Spg[k]=ws[10+k];
  float wsum = Sw + 1e-5f;
  float mu[3], mug[3];
  for (int k=0;k<3;k++){ mu[k]=Sg[k]/wsum; mug[k]=Sp[k]/wsum; }
  float H[9];
  for (int i=0;i<3;i++) for (int j=0;j<3;j++)
    H[i*3+j] = Spg[i*3+j] - mug[i]*Sg[j] - mu[j]*Sp[i] + Sw*mug[i]*mu[j];
  // A = H^T H (symmetric PSD)
  float A[9];
  for (int i=0;i<3;i++) for (int j=0;j<3;j++){
    float s=0.f; for (int k=0;k<3;k++) s += H[k*3+i]*H[k*3+j];
    A[i*3+j]=s;
  }
  float V[9] = {1,0,0, 0,1,0, 0,0,1};
  for (int sweep=0; sweep<12; ++sweep){
    for (int pi=0; pi<3; ++pi){
      int p = (pi==2)?1:0;
      int q = (pi==0)?1:2;
      float apq = A[p*3+q];
      if (fabsf(apq) < 1e-20f) continue;
      float app=A[p*3+p], aqq=A[q*3+q];
      float th = 0.5f*(aqq-app)/apq;
      float tj = ((th>=0.f)?1.f:-1.f)/(fabsf(th)+sqrtf(th*th+1.f));
      float cj = 1.0f/sqrtf(tj*tj+1.f);
      float sj = tj*cj;
      for (int k=0;k<3;k++){
        float akp=A[k*3+p], akq=A[k*3+q];
        A[k*3+p]=cj*akp - sj*akq; A[k*3+q]=sj*akp + cj*akq;
      }
      for (int k=0;k<3;k++){
        float apk=A[p*3+k], aqk=A[q*3+k];
        A[p*3+k]=cj*apk - sj*aqk; A[q*3+k]=sj*apk + cj*aqk;
      }
      for (int k=0;k<3;k++){
        float vkp=V[k*3+p], vkq=V[k*3+q];
        V[k*3+p]=cj*vkp - sj*vkq; V[k*3+q]=sj*vkp + cj*vkq;
      }
    }
  }
  float lam[3]={A[0],A[4],A[8]};
  int ord[3]={0,1,2};
  for (int i=0;i<2;i++) for (int j=i+1;j<3;j++)
    if (lam[ord[j]] > lam[ord[i]]) { int tmp=ord[i]; ord[i]=ord[j]; ord[j]=tmp; }
  float Vs[9];
  for (int c=0;c<3;c++) for (int r=0;r<3;r++) Vs[r*3+c]=V[r*3+ord[c]];
  float U[9];
  for (int c=0;c<3;c++){
    float u[3];
    for (int r=0;r<3;r++){
      float s=0.f; for (int k=0;k<3;k++) s += H[r*3+k]*Vs[k*3+c];
      u[r]=s;
    }
    float n = sqrtf(u[0]*u[0]+u[1]*u[1]+u[2]*u[2]);
    if (n > 1e-12f){ u[0]/=n; u[1]/=n; u[2]/=n; }
    else if (c==2){
      // degenerate smallest singular: complete right-handed basis
      float c0x=U[0], c0y=U[3], c0z=U[6];
      float c1x=U[1], c1y=U[4], c1z=U[7];
      u[0]=c0y*c1z-c0z*c1y; u[1]=c0z*c1x-c0x*c1z; u[2]=c0x*c1y-c0y*c1x;
      float nn = sqrtf(u[0]*u[0]+u[1]*u[1]+u[2]*u[2]);
      if (nn > 1e-12f){ u[0]/=nn; u[1]/=nn; u[2]/=nn; } else { u[0]=0;u[1]=0;u[2]=1; }
    }
    U[0*3+c]=u[0]; U[1*3+c]=u[1]; U[2*3+c]=u[2];
  }
  float detU = U[0]*(U[4]*U[8]-U[5]*U[7]) - U[1]*(U[3]*U[8]-U[5]*U[6]) + U[2]*(U[3]*U[7]-U[4]*U[6]);
  float detV = Vs[0]*(Vs[4]*Vs[8]-Vs[5]*Vs[7]) - Vs[1]*(Vs[3]*Vs[8]-Vs[5]*Vs[6]) + Vs[2]*(Vs[3]*Vs[7]-Vs[4]*Vs[6]);
  if (detU*detV < 0.f){ U[2]=-U[2]; U[5]=-U[5]; U[8]=-U[8]; }   // flip U[:, -1]
  for (int i=0;i<3;i++) for (int j=0;j<3;j++){
    float s=0.f; for (int k=0;k<3;k++) s += U[i*3+k]*Vs[j*3+k]; // R = U V^T
    ws[19+i*3+j]=s;
  }
  ws[28]=mu[0]; ws[29]=mu[1]; ws[30]=mu[2];
  ws[31]=mug[0]; ws[32]=mug[1]; ws[33]=mug[2];
}

// ------------------------------------------------------------------
// MSE against rigidly-aligned gt
// ------------------------------------------------------------------
__global__ __launch_bounds__(256)
void k_mse(const float* __restrict__ pred, const float* __restrict__ gt,
           const float* __restrict__ ae, float* __restrict__ ws)
{
  __shared__ float red[256];
  const int t = threadIdx.x;
  const int a = blockIdx.x*256 + t;
  float R[9]; for (int k=0;k<9;k++) R[k]=ws[19+k];
  float mu0=ws[28], mu1=ws[29], mu2=ws[30];
  float mg0=ws[31], mg1=ws[32], mg2=ws[33];
  float m = ae[a];
  float g0=gt[a*3+0]-mu0, g1=gt[a*3+1]-mu1, g2=gt[a*3+2]-mu2;
  float al0 = R[0]*g0 + R[1]*g1 + R[2]*g2 + mg0;
  float al1 = R[3]*g0 + R[4]*g1 + R[5]*g2 + mg1;
  float al2 = R[6]*g0 + R[7]*g1 + R[8]*g2 + mg2;
  float d0 = pred[a*3+0]-al0, d1 = pred[a*3+1]-al1, d2 = pred[a*3+2]-al2;
  float e = (d0*d0 + d1*d1 + d2*d2 + 1e-5f) * m * m;   // *mask*weights

  red[t]=e; __syncthreads();
  for (int st=128; st>0; st>>=1){ if (t<st) red[t]+=red[t+st]; __syncthreads(); }
  if (t==0) atomicAdd(&ws[34], red[0]);
  __syncthreads();
  red[t]=m; __syncthreads();
  for (int st=128; st>0; st>>=1){ if (t<st) red[t]+=red[t+st]; __syncthreads(); }
  if (t==0) atomicAdd(&ws[35], red[0]);
}

// ------------------------------------------------------------------
// Combine
// ------------------------------------------------------------------
__global__ void k_final(const float* __restrict__ tm, const float* __restrict__ ts,
                        const float* __restrict__ ws, float* __restrict__ out)
{
  if (threadIdx.x != 0 || blockIdx.x != 0) return;
  float S=0.f;
  for (int i=0;i<NT;i++) S += tm[i];
  float l_disto = ws[0] / (1e-6f + S*S);
  float l_lddt  = 1.f - ws[1]/(ws[2] + 1e-5f);
  float t2 = ts[0]*ts[0];
  float scale = (t2 + 256.f) / (t2*256.f + 1e-5f);   // SD_DATA = 16
  float l_mse = scale * ws[34] / (1e-5f + ws[35]) / 3.f;
  out[0] = 0.03f*l_disto + 1.0f*l_lddt + 4.0f*l_mse;
}

extern "C" void kernel_launch(void* const* d_in, const int* in_sizes, int n_in,
                              void* d_out, int out_size, void* d_ws, size_t ws_size,
                              hipStream_t stream)
{
  const float* logits = (const float*)d_in[0];   // (1,768,768,64)
  const float* pos    = (const float*)d_in[1];   // (1,768,4,3)
  const float* tmask  = (const float*)d_in[2];   // (1,768)
  const float* pred   = (const float*)d_in[3];   // (1,3072,3) denoised
  const float* gt     = (const float*)d_in[4];   // (1,3072,3) augmented gt
  const float* tsteps = (const float*)d_in[5];   // (1,1)
  const float* ae     = (const float*)d_in[6];   // (1,3072)
  float* out = (float*)d_out;
  float* ws  = (float*)d_ws;

  hipMemsetAsync(d_ws, 0, 64*sizeof(float), stream);

  k_disto <<<dim3(NT/16, NT/16), 256, 0, stream>>>(logits, pos, tmask, ws);
  k_lddt  <<<dim3(NA/16, NA/16), 256, 0, stream>>>(pred, gt, ae, ws);
  k_astats<<<NA/256, 256, 0, stream>>>(pred, gt, ae, ws);
  k_svd   <<<1, 32, 0, stream>>>(ws);
  k_mse   <<<NA/256, 256, 0, stream>>>(pred, gt, ae, ws);
  k_final <<<1, 32, 0, stream>>>(tmask, tsteps, ws, out);
}